// GraphPool_76124000354794
// MI455X (gfx1250) — compile-verified
//
#include <hip/hip_runtime.h>
#include <hip/hip_bf16.h>
#include <stdint.h>

// Problem constants (from reference)
#define B_   32
#define N_   2048
#define D_   128
#define NS_  2032   // NUM_SUPPORTS
#define NQ_  16     // NUM_QUERIES
#define KK_  1016   // K_KEEP
#define MM_  1032   // M = K_KEEP + NUM_QUERIES
#define ROWTILES_ 129            // MM_/8 row-tiles per batch (129*8 == 1032)
#define ROWS_PER_WG_ 8

typedef __attribute__((ext_vector_type(2))) float v2f;
typedef __attribute__((ext_vector_type(4))) float v4f;
typedef __attribute__((ext_vector_type(8))) float v8f;

#if defined(__has_builtin) && __has_builtin(__builtin_amdgcn_s_wait_asynccnt)
#define GP_WAIT_ASYNC(n) __builtin_amdgcn_s_wait_asynccnt(n)
#else
#define GP_WAIT_ASYNC(n) asm volatile("s_wait_asynccnt %0" :: "i"(n) : "memory")
#endif

__device__ __forceinline__ float gp_sigmoid(float x) {
  return 1.0f / (1.0f + __expf(-x));
}

// ---------------------------------------------------------------------------
// Phase 1: scores[b*N+n] = sigmoid((X[row,:]·w + bias) * 0.01)
// One wave computes 16 rows via 32 chained V_WMMA_F32_16X16X4_F32 ops.
// B-matrix columns are all identical (= w chunk), so every output column of
// the 16x16 D tile holds the same dot product; lanes 0 and 16 carry the full
// 8-row result vectors (rows 0-7 / 8-15) and store them with b128 stores.
// ---------------------------------------------------------------------------
__global__ __launch_bounds__(256) void gp_scores(const float* __restrict__ X,
                                                 const float* __restrict__ w,
                                                 const float* __restrict__ bias_p,
                                                 float* __restrict__ scores) {
  const int lane = threadIdx.x & 31;
  const int wave = (blockIdx.x * blockDim.x + threadIdx.x) >> 5;
  const int l16  = lane & 15;
  const int half = lane >> 4;              // 0 -> K pair {0,1}, 1 -> K pair {2,3}
  const long row0 = (long)wave * 16;       // 16 flattened rows (b*N + n) per wave
  const float* xrow = X + (row0 + l16) * (long)D_;
  const int koff = half * 2;

  v8f acc = {};
  for (int k0 = 0; k0 < D_; k0 += 4) {
    // A 16x4 f32 tile: v0 = K={0|2}, v1 = K={1|3} per ISA layout
    v2f a  = *(const v2f*)(xrow + k0 + koff);
    // B 4x16 f32 tile with every column = w[k0..k0+3]
    v2f bb = *(const v2f*)(w + k0 + koff);
    acc = __builtin_amdgcn_wmma_f32_16x16x4_f32(
        /*neg_a=*/false, a, /*neg_b=*/false, bb,
        /*c_mod=*/(short)0, acc, /*reuse_a=*/false, /*reuse_b=*/false);
  }

  const float bias = bias_p[0];
  float r[8];
#pragma unroll
  for (int m = 0; m < 8; ++m) r[m] = gp_sigmoid((acc[m] + bias) * 0.01f);

  // Lane 0 holds rows 0-7 (col 0), lane 16 holds rows 8-15 (col 0).
  if (l16 == 0) {
    float* op = scores + row0 + half * 8;
    v4f lo = {r[0], r[1], r[2], r[3]};
    v4f hi = {r[4], r[5], r[6], r[7]};
    *(v4f*)(op)     = lo;
    *(v4f*)(op + 4) = hi;
  }
}

// ---------------------------------------------------------------------------
// Phase 2: per batch, rank-select the KK_ smallest support scores (ascending,
// ties by lower index — matches jax.lax.top_k(-intra)). mean-subtraction is a
// monotone shift so it never changes the selection. 8 WGs per batch, each WG
// ranks 256 elements against the full score list held in LDS (broadcast reads).
// Also emits the query tail and writes idx (as float) to d_out.
// ---------------------------------------------------------------------------
__global__ __launch_bounds__(256) void gp_select(const float* __restrict__ scores,
                                                 int* __restrict__ wsIdx,
                                                 float* __restrict__ wsVal,
                                                 float* __restrict__ outIdx) {
  __shared__ float s[N_];
  const int b     = blockIdx.x >> 3;
  const int chunk = blockIdx.x & 7;
  const float* sb = scores + (long)b * N_;
  for (int i = threadIdx.x; i < N_; i += 256) s[i] = sb[i];
  __syncthreads();

  const int e = chunk * 256 + threadIdx.x;   // covers [0, 2048) exactly
  if (e < NS_) {
    const float v = s[e];
    int rank = 0;
#pragma unroll 4
    for (int j = 0; j < NS_; ++j) {
      const float sj = s[j];
      rank += (sj < v) || (sj == v && j < e);
    }
    if (rank < KK_) {
      wsIdx[b * MM_ + rank]  = e;
      wsVal[b * MM_ + rank]  = v;
      outIdx[b * MM_ + rank] = (float)e;
    }
  } else {
    const int q = e - NS_;                   // query tail: fixed indices
    wsIdx[b * MM_ + KK_ + q]  = e;
    wsVal[b * MM_ + KK_ + q]  = s[e];
    outIdx[b * MM_ + KK_ + q] = (float)e;
  }
}

// ---------------------------------------------------------------------------
// Phase 3: new_X[b,i,:] = X[b, idx[b,i], :] * val[b,i]   (one 128-wide row/WG)
// ---------------------------------------------------------------------------
__global__ __launch_bounds__(128) void gp_newx(const float* __restrict__ X,
                                               const int* __restrict__ wsIdx,
                                               const float* __restrict__ wsVal,
                                               float* __restrict__ outX) {
  const int bi = blockIdx.x;                 // over B_*MM_
  const int b  = bi / MM_;
  const int r  = wsIdx[bi];
  const float v = wsVal[bi];
  const float* src = X + ((long)b * N_ + r) * D_;
  outX[(long)bi * D_ + threadIdx.x] = src[threadIdx.x] * v;
}

// ---------------------------------------------------------------------------
// Phase 4: new_A[b,i,j] = A[b, idx[b,i], idx[b,j]]
// Per WG: cache idx[b,:] in LDS, then for 8 output rows stage the 8 KB source
// row into LDS via async global->LDS b128 copies (double-buffered, ASYNCcnt),
// gather the permuted columns out of LDS, and store coalesced.
// ---------------------------------------------------------------------------
__device__ __forceinline__ void gp_async_row(const float* gsrc, float* lbuf, int t) {
  // 256 threads x 2 x 16B = 8192B = one full row of A
  uint64_t g0 = (uint64_t)(uintptr_t)(gsrc + t * 4);
  uint64_t g1 = (uint64_t)(uintptr_t)(gsrc + 1024 + t * 4);
  uint32_t l0 = (uint32_t)(uintptr_t)(lbuf + t * 4);
  uint32_t l1 = (uint32_t)(uintptr_t)(lbuf + 1024 + t * 4);
  asm volatile("global_load_async_to_lds_b128 %0, %1, off" :: "v"(l0), "v"(g0) : "memory");
  asm volatile("global_load_async_to_lds_b128 %0, %1, off" :: "v"(l1), "v"(g1) : "memory");
}

__global__ __launch_bounds__(256) void gp_newa(const float* __restrict__ A,
                                               const int* __restrict__ wsIdx,
                                               float* __restrict__ outA) {
  __shared__ __align__(16) int   cidx[MM_];
  __shared__ __align__(16) float rowbuf[2][N_];

  const int b    = blockIdx.x / ROWTILES_;
  const int tile = blockIdx.x % ROWTILES_;
  const int i0   = tile * ROWS_PER_WG_;
  const int t    = threadIdx.x;

  const int* idxb = wsIdx + b * MM_;
  for (int i = t; i < MM_; i += 256) cidx[i] = idxb[i];
  __syncthreads();

  const float* Ab = A + (long)b * N_ * N_;

  // prefetch row 0 (row index broadcast-read from LDS)
  gp_async_row(Ab + (long)cidx[i0] * N_, rowbuf[0], t);

  for (int rr = 0; rr < ROWS_PER_WG_; ++rr) {
    if (rr + 1 < ROWS_PER_WG_) {
      gp_async_row(Ab + (long)cidx[i0 + rr + 1] * N_, rowbuf[(rr + 1) & 1], t);
      GP_WAIT_ASYNC(2);   // row rr's 2 copies done; row rr+1's may be in flight
    } else {
      GP_WAIT_ASYNC(0);
    }
    __syncthreads();      // row rr fully resident for all waves

    const float* rb = rowbuf[rr & 1];
    const long obase = ((long)b * MM_ + (i0 + rr)) * MM_;
    for (int j = t; j < MM_; j += 256)
      outA[obase + j] = rb[cidx[j]];

    __syncthreads();      // buffer free before next iteration overwrites it
  }
}

// ---------------------------------------------------------------------------
extern "C" void kernel_launch(void* const* d_in, const int* in_sizes, int n_in,
                              void* d_out, int out_size, void* d_ws, size_t ws_size,
                              hipStream_t stream) {
  const float* A = (const float*)d_in[0];
  const float* X = (const float*)d_in[1];
  const float* w = (const float*)d_in[2];
  const float* b = (const float*)d_in[3];

  float* out    = (float*)d_out;
  float* outA   = out;                                    // B*M*M
  float* outX   = outA + (size_t)B_ * MM_ * MM_;          // B*M*D
  float* outIdx = outX + (size_t)B_ * MM_ * D_;           // B*M (as float)

  float* scores = (float*)d_ws;                           // B*N floats
  int*   wsIdx  = (int*)(scores + (size_t)B_ * N_);       // B*M ints
  float* wsVal  = (float*)(wsIdx + (size_t)B_ * MM_);     // B*M floats

  gp_scores<<<(B_ * N_ / 16) / 8, 256, 0, stream>>>(X, w, b, scores);
  gp_select<<<B_ * 8, 256, 0, stream>>>(scores, wsIdx, wsVal, outIdx);
  gp_newx  <<<B_ * MM_, 128, 0, stream>>>(X, wsIdx, wsVal, outX);
  gp_newa  <<<B_ * ROWTILES_, 256, 0, stream>>>(A, wsIdx, outA);
}